// GFlownet_5918464933902
// MI455X (gfx1250) — compile-verified
//
#include <hip/hip_runtime.h>
#include <math.h>

// ---------------------------------------------------------------------------
// CDNA5 WMMA f32 16x16x4:  A = 16x4 f32 (2 VGPR/lane), B = 4x16 f32,
// C/D = 16x16 f32 (8 VGPR). Wave32.
// ---------------------------------------------------------------------------
typedef __attribute__((ext_vector_type(2))) float v2f;
typedef __attribute__((ext_vector_type(8))) float v8f;

#define SLOPE 0.01f

__device__ __forceinline__ float leaky(float v) { return v > 0.0f ? v : SLOPE * v; }

// ---------------------------------------------------------------------------
// WMMA GEMM:  Y[M,N] = act( X[idx][M,K] @ W^T + b ),  W is [N,K] row-major.
// One wave computes a 16 x (NT*16) tile; A fragment reused across NT B tiles.
// Branchless K-loop, software-pipelined: fragments for step k+4 are loaded
// before the WMMAs of step k, so the matrix ops cover the load latency.
// Out-of-range B columns are clamped to a valid address (their garbage
// results land only in masked-out output columns).
// ---------------------------------------------------------------------------
template<int NT>
__global__ void gemm_wmma_kernel(const float* __restrict__ X,
                                 const int* __restrict__ xidx,
                                 const float* __restrict__ W,
                                 const float* __restrict__ bias,
                                 float* __restrict__ Y,
                                 int M, int N, int K, int act, int ldy) {
    const int wave  = (blockIdx.x * blockDim.x + threadIdx.x) >> 5;
    const int ng    = (N + NT * 16 - 1) / (NT * 16);
    const int tiles = (M >> 4) * ng;
    if (wave >= tiles) return;                 // wave-uniform exit: EXEC stays all-1s

    const int mt  = wave / ng;
    const int ntg = wave - mt * ng;
    const int m0  = mt << 4;
    const int n0  = ntg * (NT * 16);

    const int lane = threadIdx.x & 31;
    const int half = lane >> 4;                // 0: lanes 0-15, 1: lanes 16-31
    const int l15  = lane & 15;
    const int koff = half * 2;                 // which K pair this lane holds

    // A row pointer (optionally gathered)
    int arow = m0 + l15;
    if (xidx) arow = xidx[arow];
    const float* __restrict__ xrow = X + (size_t)arow * (size_t)K + koff;

    // B row pointers (clamped columns keep addresses valid; results masked later)
    const float* wrow[NT];
    int          bcol[NT];
    v8f          c[NT];
#pragma unroll
    for (int t = 0; t < NT; ++t) {
        bcol[t] = n0 + t * 16 + l15;
        const int bc = bcol[t] < N ? bcol[t] : (N - 1);
        wrow[t] = W + (size_t)bc * (size_t)K + koff;
        const float bv = (bias && bcol[t] < N) ? bias[bcol[t]] : 0.0f;
#pragma unroll
        for (int r = 0; r < 8; ++r) c[t][r] = bv;
    }

    // ---- software-pipelined K loop ----
    v2f a0, b0[NT];
    a0.x = xrow[0];
    a0.y = xrow[1];
#pragma unroll
    for (int t = 0; t < NT; ++t) {
        b0[t].x = wrow[t][0];
        b0[t].y = wrow[t][1];
    }

#pragma unroll 2
    for (int k = 4; k < K; k += 4) {
        v2f a1, b1[NT];
        a1.x = xrow[k];
        a1.y = xrow[k + 1];
#pragma unroll
        for (int t = 0; t < NT; ++t) {
            b1[t].x = wrow[t][k];
            b1[t].y = wrow[t][k + 1];
        }
#pragma unroll
        for (int t = 0; t < NT; ++t) {
            c[t] = __builtin_amdgcn_wmma_f32_16x16x4_f32(false, a0, false, b0[t],
                                                         (short)0, c[t], false, false);
        }
        a0 = a1;
#pragma unroll
        for (int t = 0; t < NT; ++t) b0[t] = b1[t];
    }
#pragma unroll
    for (int t = 0; t < NT; ++t) {
        c[t] = __builtin_amdgcn_wmma_f32_16x16x4_f32(false, a0, false, b0[t],
                                                     (short)0, c[t], false, false);
    }

    // ---- store (ragged-N masked) ----
#pragma unroll
    for (int t = 0; t < NT; ++t) {
        if (bcol[t] < N) {
#pragma unroll
            for (int r = 0; r < 8; ++r) {
                const int row = m0 + half * 8 + r;
                float v = c[t][r];
                if (act) v = leaky(v);
                Y[(size_t)row * (size_t)ldy + bcol[t]] = v;
            }
        }
    }
}

// 128x128 transpose (conv_root -> torch [N,K] layout), run once.
__global__ void transpose128_kernel(const float* __restrict__ A,
                                    float* __restrict__ At) {
    int i = blockIdx.x * blockDim.x + threadIdx.x;
    if (i >= 128 * 128) return;
    int r = i >> 7, cl = i & 127;
    At[cl * 128 + r] = A[i];
}

// ---------------------------------------------------------------------------
// Rank-1 NNConv edge messages: msg_e = (h[src] . bond0) * bond1, scatter-add
// into agg[dst].  One wave per edge, 4 floats per lane (D=128).
// ---------------------------------------------------------------------------
__global__ void edge_msg_kernel(const float* __restrict__ h,
                                const int* __restrict__ eidx,   // [2,E]
                                const int* __restrict__ eattr,  // [E,2]
                                const float* __restrict__ emb_bond,
                                float* __restrict__ agg, int E) {
    const int e    = (blockIdx.x * blockDim.x + threadIdx.x) >> 5;
    const int lane = threadIdx.x & 31;
    if (e >= E) return;

    const int src = eidx[e];
    const int dst = eidx[E + e];
    const int t0  = eattr[2 * e];
    const int t1  = eattr[2 * e + 1];

    const float4 hs = ((const float4*)(h        + (size_t)src * 128))[lane];
    const float4 b0 = ((const float4*)(emb_bond + (size_t)t0  * 128))[lane];
    const float4 b1 = ((const float4*)(emb_bond + (size_t)t1  * 128))[lane];

    float p = hs.x * b0.x + hs.y * b0.y + hs.z * b0.z + hs.w * b0.w;
#pragma unroll
    for (int off = 16; off; off >>= 1) p += __shfl_xor(p, off, 32);

    float* ap = agg + (size_t)dst * 128 + lane * 4;
    atomicAdd(ap + 0, p * b1.x);
    atomicAdd(ap + 1, p * b1.y);
    atomicAdd(ap + 2, p * b1.z);
    atomicAdd(ap + 3, p * b1.w);
}

__global__ void fill_kernel(float* __restrict__ p, int n) {
    int i = blockIdx.x * blockDim.x + threadIdx.x;
    if (i < n) p[i] = 0.0f;
}

__global__ void deg_kernel(const int* __restrict__ eidx, float* __restrict__ deg, int E) {
    int e = blockIdx.x * blockDim.x + threadIdx.x;
    if (e < E) atomicAdd(&deg[eidx[E + e]], 1.0f);
}

__global__ void invdeg_kernel(const float* __restrict__ deg, float* __restrict__ invd, int n) {
    int i = blockIdx.x * blockDim.x + threadIdx.x;
    if (i < n) { float d = deg[i]; invd[i] = d > 0.0f ? 1.0f / d : 0.0f; }
}

// m = leaky( agg*inv_deg + root + conv_bias )
__global__ void node_m_kernel(const float* __restrict__ agg,
                              const float* __restrict__ invd,
                              const float* __restrict__ root,
                              const float* __restrict__ bias,
                              float* __restrict__ m, int total) {
    int i = blockIdx.x * blockDim.x + threadIdx.x;
    if (i >= total) return;
    int row = i >> 7, col = i & 127;
    m[i] = leaky(agg[i] * invd[row] + root[i] + bias[col]);
}

// GRU cell (torch gate order r,z,n); h updated in place.
__global__ void gru_kernel(const float* __restrict__ gi,
                           const float* __restrict__ gh,
                           float* __restrict__ h, int total) {
    int i = blockIdx.x * blockDim.x + threadIdx.x;
    if (i >= total) return;
    int row = i >> 7, col = i & 127;
    const float* gir = gi + (size_t)row * 384;
    const float* ghr = gh + (size_t)row * 384;
    float r = 1.0f / (1.0f + __expf(-(gir[col] + ghr[col])));
    float z = 1.0f / (1.0f + __expf(-(gir[128 + col] + ghr[128 + col])));
    float n = tanhf(gir[256 + col] + r * ghr[256 + col]);
    h[i] = (1.0f - z) * n + z * h[i];
}

// stem_cat[s] = concat( h[stem_node_idx[s]], emb_stem[stemtypes[s]] )
__global__ void stemcat_kernel(const float* __restrict__ h,
                               const float* __restrict__ emb_stem,
                               const int* __restrict__ snode,
                               const int* __restrict__ stype,
                               float* __restrict__ scat, int S) {
    int i = blockIdx.x * blockDim.x + threadIdx.x;
    if (i >= S * 256) return;
    int s = i >> 8, c = i & 255;
    float v = (c < 128) ? h[(size_t)snode[s] * 128 + c]
                        : emb_stem[(size_t)stype[s] * 128 + (c - 128)];
    scat[i] = v;
}

__global__ void pool_sum_kernel(const float* __restrict__ h,
                                const int* __restrict__ nbatch,
                                float* __restrict__ gsum,
                                float* __restrict__ gcnt, int total) {
    int i = blockIdx.x * blockDim.x + threadIdx.x;
    if (i >= total) return;
    int row = i >> 7, col = i & 127;
    int g = nbatch[row];
    atomicAdd(&gsum[(size_t)g * 128 + col], h[i]);
    if (col == 0) atomicAdd(&gcnt[g], 1.0f);
}

__global__ void pool_mean_kernel(const float* __restrict__ gsum,
                                 const float* __restrict__ gcnt,
                                 float* __restrict__ gmean, int total) {
    int i = blockIdx.x * blockDim.x + threadIdx.x;
    if (i >= total) return;
    gmean[i] = gsum[i] / fmaxf(gcnt[i >> 7], 1.0f);
}

// ---------------------------------------------------------------------------
// Host-side orchestration (all launches on `stream`; graph-capture safe).
// ---------------------------------------------------------------------------
static inline void gemm(const float* X, const int* xidx, const float* W,
                        const float* bias, float* Y, int M, int N, int K,
                        int act, int ldy, hipStream_t stream) {
    int ng = (N + 63) >> 6;                       // NT=4 -> 64 cols per wave
    int tiles = (M >> 4) * ng;
    int blocks = (tiles + 3) / 4;                 // 4 waves / 128-thread block
    gemm_wmma_kernel<4><<<blocks, 128, 0, stream>>>(X, xidx, W, bias, Y,
                                                    M, N, K, act, ldy);
}

extern "C" void kernel_launch(void* const* d_in, const int* in_sizes, int n_in,
                              void* d_out, int out_size, void* d_ws, size_t ws_size,
                              hipStream_t stream) {
    (void)in_sizes; (void)n_in; (void)out_size; (void)ws_size;
    const int NN = 8192, E = 16384, S = 4096, G = 256, D = 128;

    const int*   x        = (const int*)d_in[0];
    const int*   stypes   = (const int*)d_in[1];
    const int*   eattr    = (const int*)d_in[2];
    const int*   eidx     = (const int*)d_in[3];
    const int*   snode    = (const int*)d_in[4];
    const int*   nbatch   = (const int*)d_in[5];
    const float* emb_block= (const float*)d_in[6];
    const float* emb_stem = (const float*)d_in[7];
    const float* emb_bond = (const float*)d_in[8];
    const float* b2e_w1   = (const float*)d_in[9];
    const float* b2e_b1   = (const float*)d_in[10];
    const float* b2e_w2   = (const float*)d_in[11];
    const float* b2e_b2   = (const float*)d_in[12];
    const float* conv_root= (const float*)d_in[13];
    const float* conv_bias= (const float*)d_in[14];
    const float* gru_w_ih = (const float*)d_in[15];
    const float* gru_w_hh = (const float*)d_in[16];
    const float* gru_b_ih = (const float*)d_in[17];
    const float* gru_b_hh = (const float*)d_in[18];
    const float* s2p_w1   = (const float*)d_in[19];
    const float* s2p_b1   = (const float*)d_in[20];
    const float* s2p_w2   = (const float*)d_in[21];
    const float* s2p_b2   = (const float*)d_in[22];
    const float* s2p_w3   = (const float*)d_in[23];
    const float* s2p_b3   = (const float*)d_in[24];
    const float* g2p_w1   = (const float*)d_in[25];
    const float* g2p_b1   = (const float*)d_in[26];
    const float* g2p_w2   = (const float*)d_in[27];
    const float* g2p_b2   = (const float*)d_in[28];

    // Workspace layout (floats)
    float* f     = (float*)d_ws;
    float* h     = f; f += (size_t)NN * D;         // hidden state == out
    float* root  = f; f += (size_t)NN * D;         // root term
    float* agg   = f; f += (size_t)NN * D;         // edge aggregation
    float* mbuf  = f; f += (size_t)NN * D;         // conv output m (also t0, g1)
    float* gi    = f; f += (size_t)NN * 3 * D;     // GRU input gates (also scat/s1)
    float* gh    = f; f += (size_t)NN * 3 * D;     // GRU hidden gates (also s2)
    float* rootT = f; f += (size_t)D * D;          // transposed conv_root
    float* deg   = f; f += NN;
    float* invd  = f; f += NN;
    // time-separated aliases (post conv loop)
    float* scat  = gi;                 // [S, 256]
    float* s1    = gi + (size_t)S * 256;
    float* s2    = gh;
    float* gsum  = agg;                // [G, 128]
    float* gcnt  = agg + (size_t)G * D;
    float* gmean = agg + (size_t)G * D + G;

    float* stem_pred = (float*)d_out;                    // [S, 105]
    float* stop_pred = (float*)d_out + (size_t)S * 105;  // [G, 1]

    const int TPB = 256;
    const int totND = NN * D;

    // ---- one-time prep: transpose conv_root to [N,K] layout
    transpose128_kernel<<<(D * D + TPB - 1) / TPB, TPB, 0, stream>>>(conv_root, rootT);

    // ---- block2emb MLP: h = (act(emb_block[x] @ w1^T + b1)) @ w2^T + b2
    gemm(emb_block, x, b2e_w1, b2e_b1, mbuf, NN, D, D, 1, D, stream);
    gemm(mbuf, nullptr, b2e_w2, b2e_b2, h, NN, D, D, 0, D, stream);

    // ---- degrees (constant across steps)
    fill_kernel<<<(NN + TPB - 1) / TPB, TPB, 0, stream>>>(deg, NN);
    deg_kernel<<<(E + TPB - 1) / TPB, TPB, 0, stream>>>(eidx, deg, E);
    invdeg_kernel<<<(NN + TPB - 1) / TPB, TPB, 0, stream>>>(deg, invd, NN);

    // ---- 12 conv + GRU steps
    for (int it = 0; it < 12; ++it) {
        fill_kernel<<<(totND + TPB - 1) / TPB, TPB, 0, stream>>>(agg, totND);
        edge_msg_kernel<<<(E * 32 + TPB - 1) / TPB, TPB, 0, stream>>>(
            h, eidx, eattr, emb_bond, agg, E);
        gemm(h, nullptr, rootT, nullptr, root, NN, D, D, 0, D, stream);
        node_m_kernel<<<(totND + TPB - 1) / TPB, TPB, 0, stream>>>(
            agg, invd, root, conv_bias, mbuf, totND);
        gemm(mbuf, nullptr, gru_w_ih, gru_b_ih, gi, NN, 3 * D, D, 0, 3 * D, stream);
        gemm(h,    nullptr, gru_w_hh, gru_b_hh, gh, NN, 3 * D, D, 0, 3 * D, stream);
        gru_kernel<<<(totND + TPB - 1) / TPB, TPB, 0, stream>>>(gi, gh, h, totND);
    }

    // ---- stem head
    stemcat_kernel<<<(S * 256 + TPB - 1) / TPB, TPB, 0, stream>>>(
        h, emb_stem, snode, stypes, scat, S);
    gemm(scat, nullptr, s2p_w1, s2p_b1, s1, S, D, 2 * D, 1, D, stream);
    gemm(s1,   nullptr, s2p_w2, s2p_b2, s2, S, D, D, 1, D, stream);
    gemm(s2,   nullptr, s2p_w3, s2p_b3, stem_pred, S, 105, D, 0, 105, stream);

    // ---- global mean pool + stop head
    fill_kernel<<<(G * D + G + TPB - 1) / TPB, TPB, 0, stream>>>(gsum, G * D + G);
    pool_sum_kernel<<<(totND + TPB - 1) / TPB, TPB, 0, stream>>>(h, nbatch, gsum, gcnt, totND);
    pool_mean_kernel<<<(G * D + TPB - 1) / TPB, TPB, 0, stream>>>(gsum, gcnt, gmean, G * D);
    gemm(gmean, nullptr, g2p_w1, g2p_b1, mbuf, G, D, D, 1, D, stream);
    gemm(mbuf,  nullptr, g2p_w2, g2p_b2, stop_pred, G, 1, D, 0, 1, stream);
}